// ConvolutionalSelfAttention_42021960024540
// MI455X (gfx1250) — compile-verified
//
#include <hip/hip_runtime.h>
#include <hip/hip_bf16.h>
#include <stddef.h>

// Problem constants (from reference): B=4, T=4096, C=2048
#define TT 4096
#define CC 2048
#define NROWS (4 * 4096)   // B*T = 16384 timestep rows

typedef __attribute__((ext_vector_type(2))) float v2f;
typedef __attribute__((ext_vector_type(4))) float v4f;
typedef __attribute__((ext_vector_type(8))) float v8f;

// ---------------------------------------------------------------------------
// ws layout (floats):
//   [0      , 2048 )  wa[c]  = sum_o W_attn[o][c]
//   [2048   , 4096 )  wp[o]  = sum_i W_proj[o][i]
//   [4096   , 20480)  s[n]   = x[n,:] . wa          (n = b*T + t)
//   [20480  , 151552) partials for deterministic column reduction (64 x 2048)
// ---------------------------------------------------------------------------

// --- Kernel 1: W_attn column-sum partials (deterministic, coalesced) -------
__global__ void k_wattn_part(const float* __restrict__ Wa, float* __restrict__ part) {
    int c = blockIdx.x * 256 + threadIdx.x;   // column
    int g = blockIdx.y;                        // row group (32 rows)
    const float* p = Wa + (size_t)g * 32 * CC + c;
    float sum = 0.f;
#pragma unroll 8
    for (int o = 0; o < 32; ++o) sum += p[(size_t)o * CC];
    part[g * CC + c] = sum;
}

// --- Kernel 2: combine partials -> wa --------------------------------------
__global__ void k_wattn_comb(const float* __restrict__ part, float* __restrict__ wa) {
    int c = blockIdx.x * 256 + threadIdx.x;
    float sum = 0.f;
    for (int g = 0; g < 64; ++g) sum += part[g * CC + c];
    wa[c] = sum;
}

// --- Kernel 3: W_proj row sums -> wp (one wave32 per row) ------------------
__global__ void k_wproj(const float* __restrict__ Wp, float* __restrict__ wp) {
    int w    = threadIdx.x >> 5;
    int lane = threadIdx.x & 31;
    int o    = blockIdx.x * 8 + w;
    const float* rowp = Wp + (size_t)o * CC;
    float sum = 0.f;
#pragma unroll
    for (int i = 0; i < CC; i += 128) {           // coalesced b128 per lane
        v4f v = *(const v4f*)(rowp + i + lane * 4);
        sum += v.x + v.y + v.z + v.w;
    }
#pragma unroll
    for (int m = 16; m >= 1; m >>= 1) sum += __shfl_xor(sum, m, 32);
    if (lane == 0) wp[o] = sum;
}

// --- Kernel 4: s[n] = x[n,:] . wa via V_WMMA_F32_16X16X4_F32 ---------------
// 256 blocks x 256 threads. Block handles 64 rows = 4 tiles of 16 rows.
// Waves 2*rt, 2*rt+1 each cover K halves (1024) of row-tile rt; LDS combine.
__global__ void k_dot_wmma(const float* __restrict__ x, float* __restrict__ ws) {
    const float* wa_g = ws;          // [2048]
    float*       s_g  = ws + 4096;   // [16384]

    __shared__ float lds_wa[CC];
    __shared__ float comb[8][16];

    int tid = threadIdx.x;

    // Stage wa into LDS (coalesced b128)
    {
        v4f*       dst = (v4f*)lds_wa;
        const v4f* src = (const v4f*)wa_g;
        dst[tid]       = src[tid];
        dst[tid + 256] = src[tid + 256];
    }
    __syncthreads();

    int w    = tid >> 5;       // wave 0..7
    int lane = tid & 31;
    int rt   = w >> 1;         // row-tile 0..3
    int half = w & 1;          // K half
    int n0   = blockIdx.x * 64;
    int row  = n0 + rt * 16 + (lane & 15);

    // A-fragment layout (16x4 f32): lanes 0-15 -> K=0,1 ; lanes 16-31 -> K=2,3
    int delta = (lane < 16) ? 0 : 2;
    const float* ap = x + (size_t)row * CC + half * 1024 + delta;
    const float* bp = &lds_wa[half * 1024 + delta];   // B fragment: wa broadcast

    v8f acc0 = {0.f, 0.f, 0.f, 0.f, 0.f, 0.f, 0.f, 0.f};
    v8f acc1 = {0.f, 0.f, 0.f, 0.f, 0.f, 0.f, 0.f, 0.f};

    for (int kk = 0; kk < 1024; kk += 64) {
        __builtin_prefetch(ap + kk + 512, 0, 1);      // global_prefetch ahead
#pragma unroll
        for (int k = 0; k < 64; k += 8) {
            v2f a0 = *(const v2f*)(ap + kk + k);
            v2f a1 = *(const v2f*)(ap + kk + k + 4);
            v2f b0 = *(const v2f*)(bp + kk + k);
            v2f b1 = *(const v2f*)(bp + kk + k + 4);
            acc0 = __builtin_amdgcn_wmma_f32_16x16x4_f32(
                false, a0, false, b0, (short)0, acc0, false, false);
            acc1 = __builtin_amdgcn_wmma_f32_16x16x4_f32(
                false, a1, false, b1, (short)0, acc1, false, false);
        }
    }
    v8f acc = acc0 + acc1;

    // D layout: reg r, lanes 0-15 -> M=r (N=lane); lanes 16-31 -> M=8+r.
    // Every column N holds the same value; lane 0 / lane 16 extract.
    if (lane == 0 || lane == 16) {
        int m0 = (lane >> 4) * 8;
#pragma unroll
        for (int r = 0; r < 8; ++r) comb[w][m0 + r] = acc[r];
    }
    __syncthreads();

    if (tid < 64) {
        int rt2 = tid >> 4, m = tid & 15;
        s_g[n0 + rt2 * 16 + m] = comb[2 * rt2][m] + comb[2 * rt2 + 1][m];
    }
}

// --- Kernel 5: y = causal 3-tap mean of s; out[n][o] = y[n] * wp[o] --------
// 256 blocks x 256 threads; block writes 64 rows x 2048 cols with NT b128.
__global__ void k_out(const float* __restrict__ ws, float* __restrict__ out) {
    const float* wp = ws + 2048;
    const float* s  = ws + 4096;
    __shared__ float sh[66];

    int n0  = blockIdx.x * 64;
    int tid = threadIdx.x;

    if (tid < 66) {
        int n = n0 - 2 + tid;
        sh[tid] = (n >= 0) ? s[n] : 0.f;
    }
    __syncthreads();

    int c0 = tid * 8;
    v4f w0 = *(const v4f*)(wp + c0);
    v4f w1 = *(const v4f*)(wp + c0 + 4);

    for (int r = 0; r < 64; ++r) {
        int n = n0 + r;
        int t = n & (TT - 1);          // per-b temporal index; tiles never straddle b
        float y = sh[r + 2];
        if (t >= 1) y += sh[r + 1];
        if (t >= 2) y += sh[r];
        y *= (1.0f / 3.0f);
        float* po = out + (size_t)n * CC + c0;
        __builtin_nontemporal_store(w0 * y, (v4f*)po);
        __builtin_nontemporal_store(w1 * y, (v4f*)(po + 4));
    }
}

// ---------------------------------------------------------------------------
extern "C" void kernel_launch(void* const* d_in, const int* in_sizes, int n_in,
                              void* d_out, int out_size, void* d_ws, size_t ws_size,
                              hipStream_t stream) {
    const float* x  = (const float*)d_in[0];   // [4,4096,2048] f32
    const float* Wa = (const float*)d_in[1];   // [2048,2048]   f32
    const float* Wp = (const float*)d_in[2];   // [2048,2048]   f32
    float* out = (float*)d_out;                // [4,4096,2048] f32
    float* ws  = (float*)d_ws;

    float* wa   = ws;            // 2048
    float* wp   = ws + 2048;     // 2048
    float* part = ws + 20480;    // 64*2048

    k_wattn_part<<<dim3(8, 64), 256, 0, stream>>>(Wa, part);
    k_wattn_comb<<<8, 256, 0, stream>>>(part, wa);
    k_wproj<<<256, 256, 0, stream>>>(Wp, wp);
    k_dot_wmma<<<256, 256, 0, stream>>>(x, ws);
    k_out<<<256, 256, 0, stream>>>(ws, out);
}